// AttentionAggregator_19542101197284
// MI455X (gfx1250) — compile-verified
//
#include <hip/hip_runtime.h>
#include <math.h>

#define D 256
#define LDS_STRIDE 260   // pad: 260%64=4 -> conflict-free row access; 260*4 bytes %16==0 -> float4-aligned rows

typedef __attribute__((ext_vector_type(2))) float v2f;
typedef __attribute__((ext_vector_type(8))) float v8f;

// ---------------------------------------------------------------------------
// Kernel 1: new_emb = feature_b @ W + bias   (f32 WMMA 16x16x4)
// One block = 16 rows (M-tile). 8 waves, each wave owns two 16-wide N-tiles.
// A panel (16 x 256) staged in LDS once per block, shared by all 8 waves.
// ---------------------------------------------------------------------------
__global__ void __launch_bounds__(256)
gat_gemm_bias_wmma(const float* __restrict__ Bfeat,
                   const float* __restrict__ W,
                   const float* __restrict__ bias,
                   float* __restrict__ new_emb,
                   int NB)
{
    __shared__ float lds_a[16 * LDS_STRIDE];

    const int m0  = blockIdx.x * 16;
    const int tid = threadIdx.x;

    // ---- stage A panel: 16 rows x 256 cols of feature_b into LDS ----------
    {
        const int row = tid >> 4;          // 0..15
        const int c0  = (tid & 15) << 4;   // 0,16,...,240  (16 floats per thread)
        const int gr  = m0 + row;
        float4 x0 = {0,0,0,0}, x1 = x0, x2 = x0, x3 = x0;
        if (gr < NB) {
            const float4* s4 = (const float4*)(Bfeat + (size_t)gr * D + c0);
            x0 = s4[0]; x1 = s4[1]; x2 = s4[2]; x3 = s4[3];
        }
        float4* dst = (float4*)(lds_a + row * LDS_STRIDE + c0);
        dst[0] = x0; dst[1] = x1; dst[2] = x2; dst[3] = x3;
    }
    __syncthreads();

    const int wave = tid >> 5;       // 0..7
    const int lane = tid & 31;
    const int lo   = lane & 15;      // N index within tile / M row for A
    const int hi   = lane >> 4;      // lane-half selector
    const int n0   = wave * 32;      // this wave covers N-tiles [n0, n0+32)

    v8f c0 = {}; // tile at columns n0..n0+15
    v8f c1 = {}; // tile at columns n0+16..n0+31

    #pragma unroll 4
    for (int k = 0; k < D; k += 4) {
        // A fragment (ISA layout): lanes 0-15 hold K=k,k+1; lanes 16-31 hold K=k+2,k+3
        const float2 af = *(const float2*)(lds_a + lo * LDS_STRIDE + k + 2 * hi);
        v2f a; a.x = af.x; a.y = af.y;

        // B fragments: VGPR0 = W[k+2*hi][n], VGPR1 = W[k+1+2*hi][n]
        const int kr = k + 2 * hi;
        const float* wp0 = W + (size_t)kr * D + n0 + lo;
        v2f b0; b0.x = wp0[0];  b0.y = wp0[D];
        v2f b1; b1.x = wp0[16]; b1.y = wp0[D + 16];

        c0 = __builtin_amdgcn_wmma_f32_16x16x4_f32(false, a, false, b0, (short)0, c0, false, false);
        c1 = __builtin_amdgcn_wmma_f32_16x16x4_f32(false, a, false, b1, (short)0, c1, false, false);
    }

    // ---- bias + store. C layout: VGPR r -> (M = r + 8*hi, N = lo) ---------
    const float bv0 = bias[n0 + lo];
    const float bv1 = bias[n0 + 16 + lo];
    #pragma unroll
    for (int r = 0; r < 8; ++r) {
        const int m = m0 + r + 8 * hi;
        if (m < NB) {
            float* o = new_emb + (size_t)m * D + n0 + lo;
            o[0]  = c0[r] + bv0;
            o[16] = c1[r] + bv1;
        }
    }
}

// ---------------------------------------------------------------------------
// Kernel 2: out[i] = dot(X[i,:], vec[:])  (one wave32 per row, D=256)
// ---------------------------------------------------------------------------
__global__ void __launch_bounds__(256)
gat_rowdot(const float* __restrict__ X,
           const float* __restrict__ vec,
           float* __restrict__ out,
           int N)
{
    const int row  = (int)((blockIdx.x * blockDim.x + threadIdx.x) >> 5);
    const int lane = threadIdx.x & 31;
    if (row >= N) return;

    const float4* r4 = (const float4*)(X + (size_t)row * D + lane * 8);
    const float4* v4 = (const float4*)(vec + lane * 8);
    const float4 x0 = r4[0], x1 = r4[1];
    const float4 w0 = v4[0], w1 = v4[1];

    float s = x0.x * w0.x + x0.y * w0.y + x0.z * w0.z + x0.w * w0.w
            + x1.x * w1.x + x1.y * w1.y + x1.z * w1.z + x1.w * w1.w;

    #pragma unroll
    for (int off = 16; off >= 1; off >>= 1)
        s += __shfl_xor(s, off, 32);

    if (lane == 0) out[row] = s;
}

// ---------------------------------------------------------------------------
// Kernel 3: edge aggregation. edge_src is SORTED -> one wave32 per src node:
// binary-search edge range, accumulate 256-wide weighted sum in registers,
// fuse the softmax-style normalization into the store. No atomics.
// ---------------------------------------------------------------------------
__device__ __forceinline__ int lower_bound_i64(const long long* __restrict__ arr,
                                               int n, long long key)
{
    int lo = 0, hi = n;
    while (lo < hi) {
        int mid = (lo + hi) >> 1;
        if (arr[mid] < key) lo = mid + 1; else hi = mid;
    }
    return lo;
}

__global__ void __launch_bounds__(256)
gat_edge_aggregate(const float* __restrict__ new_emb,
                   const float* __restrict__ s_a,
                   const float* __restrict__ s_b,
                   const long long* __restrict__ esrc,
                   const long long* __restrict__ edst,
                   float* __restrict__ out,
                   int E, int NA)
{
    const int src  = (int)((blockIdx.x * blockDim.x + threadIdx.x) >> 5);
    const int lane = threadIdx.x & 31;
    if (src >= NA) return;

    const int elo = lower_bound_i64(esrc, E, (long long)src);
    const int ehi = lower_bound_i64(esrc, E, (long long)src + 1);

    const float sa  = s_a[src];
    const int   col = lane * 8;

    float a0 = 0.f, a1 = 0.f, a2 = 0.f, a3 = 0.f;
    float a4 = 0.f, a5 = 0.f, a6 = 0.f, a7 = 0.f;
    float ssum = 0.f;

    for (int e = elo; e < ehi; ++e) {
        const long long dst = edst[e];
        const float logit = sa + s_b[dst];
        const float eluv  = (logit > 0.f) ? logit : 0.1f * expm1f(logit);
        const float score = expf(eluv);
        ssum += score;

        const float4* p = (const float4*)(new_emb + (size_t)dst * D + col);
        const float4 v0 = p[0];
        const float4 v1 = p[1];
        a0 += score * v0.x; a1 += score * v0.y; a2 += score * v0.z; a3 += score * v0.w;
        a4 += score * v1.x; a5 += score * v1.y; a6 += score * v1.z; a7 += score * v1.w;
    }

    const float inv = (ssum == 0.f) ? 1.f : (1.f / ssum);
    float4 o0 = { a0 * inv, a1 * inv, a2 * inv, a3 * inv };
    float4 o1 = { a4 * inv, a5 * inv, a6 * inv, a7 * inv };
    float4* op = (float4*)(out + (size_t)src * D + col);
    op[0] = o0;
    op[1] = o1;
}

// ---------------------------------------------------------------------------
// Launch
// ---------------------------------------------------------------------------
extern "C" void kernel_launch(void* const* d_in, const int* in_sizes, int n_in,
                              void* d_out, int out_size, void* d_ws, size_t ws_size,
                              hipStream_t stream)
{
    const float*     feature_a = (const float*)d_in[0];
    const float*     feature_b = (const float*)d_in[1];
    const float*     W         = (const float*)d_in[2];
    const float*     bias      = (const float*)d_in[3];
    const float*     a_vec     = (const float*)d_in[4];   // 2*D floats
    const long long* esrc      = (const long long*)d_in[5];
    const long long* edst      = (const long long*)d_in[6];

    const int NA = in_sizes[0] / D;
    const int NB = in_sizes[1] / D;
    const int E  = in_sizes[5];

    float* ws      = (float*)d_ws;
    float* new_emb = ws;                          // NB * D floats (51.2 MB)
    float* s_a     = new_emb + (size_t)NB * D;    // NA floats
    float* s_b     = s_a + NA;                    // NB floats

    // 1) new_emb = feature_b @ W + b   (WMMA f32)
    gat_gemm_bias_wmma<<<(NB + 15) / 16, 256, 0, stream>>>(feature_b, W, bias, new_emb, NB);

    // 2) s_a = feature_a @ a_top ; s_b = new_emb @ a_bot
    gat_rowdot<<<(NA + 7) / 8, 256, 0, stream>>>(feature_a, a_vec,     s_a, NA);
    gat_rowdot<<<(NB + 7) / 8, 256, 0, stream>>>(new_emb,   a_vec + D, s_b, NB);

    // 3) per-src-wave edge aggregation with fused normalization
    gat_edge_aggregate<<<(NA + 7) / 8, 256, 0, stream>>>(new_emb, s_a, s_b,
                                                         esrc, edst, (float*)d_out, E, NA);
}